// SpectralConv1d_x_14130442404384
// MI455X (gfx1250) — compile-verified
//
#include <hip/hip_runtime.h>
#include <math.h>

// ---------------------------------------------------------------------------
// SpectralConv1d on MI455X (gfx1250), 3 WMMA GEMM stages:
//   1) XF[32, 1M]  = F[32,64]   @ x[64, 1M]          (truncated rFFT, 16 modes)
//   2) OF[m]       = XF[m]      @ Wblk[m][128,128]   (complex mix as real block GEMM)
//   3) out[64, 1M] = G[64,32]   @ OF[32-view, 1M]    (zero-padded irFFT)
// f32 WMMA 16x16x4; each wave owns multiple 16x16 tiles so B-operands are
// loaded once and reused (x and OF each read exactly once from HBM).
// ---------------------------------------------------------------------------

typedef __attribute__((ext_vector_type(2))) float v2f;
typedef __attribute__((ext_vector_type(8))) float v8f;

#define C_IN  64
#define MODES 16
#define ND    16384            // N*D = 1024*16
#define NDC   1048576          // ND*C_IN
#define NDO   1048576          // ND*C_OUT
#define CH2   128              // 2*C_IN (re|im channel block)

// workspace layout in floats
#define F_OFF   0                          // F[32][64]
#define G_OFF   2048                       // G[64][32]
#define W_OFF   4096                       // Wblk[16][128][128]
#define XF_OFF  (4096 + 16*128*128)        // XF[16][16384][128]
#define OF_OFF  (XF_OFF + 16*ND*CH2)       // OF[16][16384][128]

__device__ inline v8f wmma_f32_16x16x4(v2f a, v2f b, v8f c) {
  return __builtin_amdgcn_wmma_f32_16x16x4_f32(
      false, a, false, b, (short)0, c, false, false);
}

// ---------------------------------------------------------------------------
// Setup: DFT matrix F, iDFT matrix G, block-complex weights Wblk.
//   F[2m+ri][t] : Re = cos(th*m*t), Im = -sin(th*m*t)      (e^{-i th m t})
//   Wblk[m]     : [[Wr, Wi], [-Wi, Wr]]  so [Xr Xi] @ Wblk = [Or Oi]
//   G[t][2m+ri] : (1/T) * {m=0: (1,0); m>0: (2cos, -2sin)} (irfft drops DC Im)
// ---------------------------------------------------------------------------
__global__ void setup_kernel(const float* __restrict__ w1, float* __restrict__ ws) {
  int idx = blockIdx.x * blockDim.x + threadIdx.x;
  const float TH = 6.283185307179586f / 64.0f;
  const int WN = 16 * 128 * 128;
  if (idx < WN) {
    int m  = idx >> 14;
    int k  = (idx >> 7) & 127;
    int ch = idx & 127;
    int i  = k & 63;
    int o  = ch & 63;
    float wr = w1[((i * 64 + o) * 16 + m) * 2 + 0];
    float wi = w1[((i * 64 + o) * 16 + m) * 2 + 1];
    float v;
    if (k < 64) v = (ch < 64) ? wr : wi;
    else        v = (ch < 64) ? -wi : wr;
    ws[W_OFF + idx] = v;
  } else if (idx < WN + 2048) {
    int j = idx - WN;               // F[mch][t]
    int mch = j >> 6, t = j & 63;
    float ang = TH * (float)((mch >> 1) * t);
    ws[F_OFF + j] = (mch & 1) ? -sinf(ang) : cosf(ang);
  } else if (idx < WN + 4096) {
    int j = idx - (WN + 2048);      // G[t][mc]
    int t = j >> 5, mc = j & 31;
    int m = mc >> 1;
    float ang = TH * (float)(m * t);
    float v;
    if (m == 0) v = (mc & 1) ? 0.0f : 1.0f;
    else        v = (mc & 1) ? (-2.0f * sinf(ang)) : (2.0f * cosf(ang));
    ws[G_OFF + j] = v * (1.0f / 64.0f);
  }
}

// ---------------------------------------------------------------------------
// Stage 1: XF = F @ x.  Wave = 2 mch-tiles x 4 col-tiles (one 64-chan group),
// so each x fragment feeds 2 WMMAs and 4 tiles share one base address.
// x is read from HBM exactly once.  F staged in LDS.
// ---------------------------------------------------------------------------
__global__ void __launch_bounds__(128)
dft_kernel(const float* __restrict__ x, float* __restrict__ ws) {
  __shared__ float Fl[32 * 64];
  const float* Fg = ws + F_OFF;
  float* XF = ws + XF_OFF;
  for (int i = threadIdx.x; i < 2048; i += 128) Fl[i] = Fg[i];
  __syncthreads();

  const int lane = threadIdx.x & 31;
  const int wave = threadIdx.x >> 5;
  const int colBase = (blockIdx.x * 4 + wave) * 64;   // over nd*64+c (1M)
  const int r  = lane & 15;
  const int h2 = (lane >> 4) << 1;                    // K sub-slot 0|2
  v8f acc[2][4] = {};
  for (int kt = 0; kt < 16; ++kt) {
    const int tb = kt * 4;
    v2f a0, a1;
    a0.x = Fl[r * 64 + tb + h2];
    a0.y = Fl[r * 64 + tb + h2 + 1];
    a1.x = Fl[(r + 16) * 64 + tb + h2];
    a1.y = Fl[(r + 16) * 64 + tb + h2 + 1];
    const float* bp0 = x + (size_t)(tb + h2) * NDC + colBase + r;
    const float* bp1 = bp0 + NDC;
#pragma unroll
    for (int n = 0; n < 4; ++n) {
      v2f b;
      b.x = bp0[n * 16];
      b.y = bp1[n * 16];
      acc[0][n] = wmma_f32_16x16x4(a0, b, acc[0][n]);
      acc[1][n] = wmma_f32_16x16x4(a1, b, acc[1][n]);
    }
  }
  const int nd = colBase >> 6;
  const int rowOff = (lane >> 4) << 3;
#pragma unroll
  for (int mt = 0; mt < 2; ++mt)
#pragma unroll
    for (int n = 0; n < 4; ++n)
#pragma unroll
      for (int j = 0; j < 8; ++j) {
        const int mch = mt * 16 + rowOff + j;
        XF[(size_t)(mch >> 1) * (ND * CH2) + (size_t)nd * CH2 +
           (mch & 1) * 64 + n * 16 + r] = acc[mt][n][j];
      }
}

// ---------------------------------------------------------------------------
// Stage 2: OF[m] = XF[m] @ Wblk[m].  Workgroup = one 16-row nd strip of one
// mode.  The 8 KB XF strip is brought in with CDNA5 async global->LDS copies
// (ASYNCcnt path); 8 waves cover the 128 output channels.
// ---------------------------------------------------------------------------
__global__ void __launch_bounds__(256)
mix_kernel(float* __restrict__ ws) {
  __shared__ float lds[16 * CH2];
  const int m = blockIdx.y;
  const int ndBase = blockIdx.x * 16;
  const float* XF = ws + XF_OFF + (size_t)m * (ND * CH2);
  const float* W  = ws + W_OFF + (size_t)m * (CH2 * CH2);
  float* OF = ws + OF_OFF + (size_t)m * (ND * CH2);

  {
    // each thread async-copies 16 B at byte tid*16 and tid*16+4096
    // (INST_OFFSET applies to both the LDS and the global address)
    unsigned lds0 = (unsigned)(uintptr_t)(&lds[0]) + threadIdx.x * 16u;
    unsigned long long g =
        (unsigned long long)(uintptr_t)(XF + (size_t)ndBase * CH2 + threadIdx.x * 4);
    asm volatile(
        "global_load_async_to_lds_b128 %0, %1, off\n\t"
        "global_load_async_to_lds_b128 %0, %1, off offset:4096\n\t"
        "s_wait_asynccnt 0x0"
        :: "v"(lds0), "v"(g) : "memory");
  }
  __syncthreads();

  const int lane = threadIdx.x & 31;
  const int colBase = (threadIdx.x >> 5) * 16;       // wave -> 16 of 128 chans
  const int r  = lane & 15;
  const int h2 = (lane >> 4) << 1;
  v8f acc = {};
  for (int kt = 0; kt < 32; ++kt) {
    const int kb = kt * 4;
    v2f a, b;
    a.x = lds[r * CH2 + kb + h2];
    a.y = lds[r * CH2 + kb + h2 + 1];
    b.x = W[(kb + h2) * CH2 + colBase + r];
    b.y = W[(kb + h2 + 1) * CH2 + colBase + r];
    acc = wmma_f32_16x16x4(a, b, acc);
  }
  const int rowOff = (lane >> 4) << 3;
#pragma unroll
  for (int j = 0; j < 8; ++j)
    OF[(size_t)(ndBase + rowOff + j) * CH2 + colBase + r] = acc[j];
}

// ---------------------------------------------------------------------------
// Stage 3: out = G @ OF (K = 32).  Wave = 4 t-tiles x 2 col-tiles: every OF
// fragment is reused by all four time tiles, so OF is read exactly once.
// G staged in LDS.
// ---------------------------------------------------------------------------
__global__ void __launch_bounds__(128)
idft_kernel(const float* __restrict__ ws, float* __restrict__ out) {
  __shared__ float Gl[64 * 32];
  const float* Gg = ws + G_OFF;
  const float* OF = ws + OF_OFF;
  for (int i = threadIdx.x; i < 2048; i += 128) Gl[i] = Gg[i];
  __syncthreads();

  const int lane = threadIdx.x & 31;
  const int wave = threadIdx.x >> 5;
  const int colBase = (blockIdx.x * 4 + wave) * 32;  // over nd*64+o (1M)
  const int r  = lane & 15;
  const int h2 = (lane >> 4) << 1;
  const int nd = colBase >> 6;
  const int oB = colBase & 63;                       // 0 or 32
  v8f acc[4][2] = {};
  for (int kt = 0; kt < 8; ++kt) {
    const int kb = kt * 4;
    const int j0 = kb + h2, j1 = j0 + 1;             // j0 always even
    v2f a[4];
#pragma unroll
    for (int tt = 0; tt < 4; ++tt) {
      a[tt].x = Gl[(tt * 16 + r) * 32 + kb + h2];
      a[tt].y = Gl[(tt * 16 + r) * 32 + kb + h2 + 1];
    }
    const float* bp0 = OF + (size_t)(j0 >> 1) * (ND * CH2) + (size_t)nd * CH2 +
                       (j0 & 1) * 64 + oB + r;
    const float* bp1 = OF + (size_t)(j1 >> 1) * (ND * CH2) + (size_t)nd * CH2 +
                       (j1 & 1) * 64 + oB + r;
#pragma unroll
    for (int n = 0; n < 2; ++n) {
      v2f b;
      b.x = bp0[n * 16];
      b.y = bp1[n * 16];
#pragma unroll
      for (int tt = 0; tt < 4; ++tt)
        acc[tt][n] = wmma_f32_16x16x4(a[tt], b, acc[tt][n]);
    }
  }
  const int rowOff = (lane >> 4) << 3;
#pragma unroll
  for (int tt = 0; tt < 4; ++tt)
#pragma unroll
    for (int n = 0; n < 2; ++n)
#pragma unroll
      for (int j = 0; j < 8; ++j)
        out[(size_t)(tt * 16 + rowOff + j) * NDO + colBase + n * 16 + r] =
            acc[tt][n][j];
}

// ---------------------------------------------------------------------------
extern "C" void kernel_launch(void* const* d_in, const int* in_sizes, int n_in,
                              void* d_out, int out_size, void* d_ws, size_t ws_size,
                              hipStream_t stream) {
  const float* x  = (const float*)d_in[0];   // (64, 1024, 16, 64) f32
  const float* w1 = (const float*)d_in[1];   // (64, 64, 16, 2) f32
  float* ws  = (float*)d_ws;
  float* out = (float*)d_out;                // (64, 1024, 16, 64) f32

  const int setupN = 16 * 128 * 128 + 4096;
  setup_kernel<<<(setupN + 255) / 256, 256, 0, stream>>>(w1, ws);

  // Stage 1: 1M cols / (64 per wave) / 4 waves per block = 4096 blocks
  dft_kernel<<<dim3(NDC / 64 / 4), 128, 0, stream>>>(x, ws);

  // Stage 2: 1024 nd strips x 16 modes, 256 threads (8 waves)
  mix_kernel<<<dim3(ND / 16, MODES), 256, 0, stream>>>(ws);

  // Stage 3: 1M cols / (32 per wave) / 4 waves per block = 8192 blocks
  idft_kernel<<<dim3(NDO / 32 / 4), 128, 0, stream>>>(ws, out);
}